// SpikeLayer_78391743087294
// MI455X (gfx1250) — compile-verified
//
#include <hip/hip_runtime.h>

namespace {

constexpr int C   = 128;       // channels
constexpr int S   = 128;       // spikes
constexpr int HW  = 64 * 64;   // pixels per image
constexpr int BLK = 128;       // threads per block (pixels per block), 4 waves
constexpr int WPIX = 32;       // pixels per wave (wave32)

typedef unsigned int u32x4 __attribute__((ext_vector_type(4)));
typedef int          i32x4 __attribute__((ext_vector_type(4)));
typedef int          i32x8 __attribute__((ext_vector_type(8)));

#if defined(__has_builtin)
#if __has_builtin(__builtin_amdgcn_tensor_load_to_lds)
#define HAVE_TDM 1
#endif
#endif

__device__ __forceinline__ void wait_tensorcnt0() {
#if defined(__has_builtin) && __has_builtin(__builtin_amdgcn_s_wait_tensorcnt)
    __builtin_amdgcn_s_wait_tensorcnt(0);
#else
    asm volatile("s_wait_tensorcnt 0x0" ::: "memory");
#endif
    asm volatile("" ::: "memory");   // compiler fence: DS scan must not hoist above wait
}

__global__ __launch_bounds__(BLK)
void spike_sample_kernel(const float* __restrict__ inp,
                         const float* __restrict__ rnd,
                         float* __restrict__ out)
{
    // Compact per-wave CDF tiles: wave w owns elements [w*4096, (w+1)*4096).
    // Element (c, lane) of wave w lives at  w*4096 + c*32 + lane.
    // Bank = (c*32 + lane) % 64 -> 32 distinct banks per wave32 access: conflict-free.
    __shared__ float cdf[C * BLK];   // 64 KB -> 5 blocks / 320 KB WGP (20 waves)

    const int tid   = threadIdx.x;
    const int lane  = tid & 31;
    const int wv    = tid >> 5;
    const int wbase = wv << 12;               // wv * 4096 elements

    const int pid = blockIdx.x * BLK + tid;   // flat pixel id over B*H*W
    const int b   = pid >> 12;                // pid / 4096  (block never straddles images)
    const int p   = pid & (HW - 1);

    float run = 0.0f;
    float coarse[8];                          // cdf[15], cdf[31], ..., cdf[127]

#if defined(HAVE_TDM)
    // ---- Phase 1a: TDM DMA of this wave's 32x128 f32 tile into LDS ------
    // One tensor_load_to_lds per wave replaces 128 clause VMEM loads; each
    // wave waits only on its own TENSORcnt, so no workgroup barrier needed.
    {
        const int pblk = blockIdx.x * BLK;               // scalar pixel base
        const int bb   = pblk >> 12;
        const int pp   = pblk & (HW - 1);
        const int swv  = __builtin_amdgcn_readfirstlane(tid) >> 5;  // scalar wave id

        const unsigned long long ga =
            (unsigned long long)(const void*)(inp + bb * (C * HW) + pp + swv * WPIX);
        const unsigned int lds_byte =
            (unsigned int)(unsigned long long)(const void*)&cdf[0] + (unsigned)(swv * WPIX * C * 4);

        // D# group 0: count=1 | lds_addr | global_addr[56:0] | type=2
        u32x4 g0;
        g0.x = 1u;
        g0.y = lds_byte;
        g0.z = (unsigned int)(ga & 0xFFFFFFFFull);
        g0.w = (unsigned int)((ga >> 32) & 0x01FFFFFFull) | (2u << 30);

        // D# group 1: 2D tile, 4-byte elements, no pad / multicast / iterate
        i32x8 g1;
        g1[0] = (2 << 16);          // data_size = 2 (4 bytes)
        g1[1] = (int)(4096u << 16); // tensor_dim0 = 4096 (row length, elements)
        g1[2] = (C << 16);          // tensor_dim1 = 128 rows (channels)
        g1[3] = (WPIX << 16);       // tile_dim0  = 32 elements / row
        g1[4] = C;                  // tile_dim1  = 128 rows, tile_dim2 = 0
        g1[5] = 4096;               // tensor_dim0_stride = H*W elements
        g1[6] = 0;
        g1[7] = 0;

        const i32x4 gz = {0, 0, 0, 0};   // groups 2/3 unused (<=2D tensor)
#if defined(__clang_major__) && (__clang_major__ >= 23)
        const i32x8 gz8 = {0, 0, 0, 0, 0, 0, 0, 0};
        __builtin_amdgcn_tensor_load_to_lds(g0, g1, gz, gz, gz8, 0);
#else
        __builtin_amdgcn_tensor_load_to_lds(g0, g1, gz, gz, 0);
#endif
        wait_tensorcnt0();
    }

    // ---- Phase 1b: in-place inclusive prefix sum over LDS tile ----------
#pragma unroll
    for (int seg = 0; seg < 8; ++seg) {
#pragma unroll
        for (int k = 0; k < 16; ++k) {
            const int c = seg * 16 + k;
            const int a = wbase + c * WPIX + lane;
            run += cdf[a];
            cdf[a] = run;
        }
        coarse[seg] = run;
    }
#else
    // ---- Fallback: clause-streamed global loads + LDS stores ------------
    const float* ip = inp + b * (C * HW) + p;
#pragma unroll
    for (int seg = 0; seg < 8; ++seg) {
#pragma unroll
        for (int k = 0; k < 16; ++k) {
            const int c = seg * 16 + k;
            run += ip[c * HW];
            cdf[wbase + c * WPIX + lane] = run;
        }
        coarse[seg] = run;
    }
#endif

    const float total = run;   // == cdf[C-1]; compare r*total against raw cdf

    const float* rp = rnd + b * (S * HW) + p;
    float*       op = out + b * (S * HW) + p;

    // ---- Phase 2: searchsorted(left) per spike --------------------------
#pragma unroll 4
    for (int s = 0; s < S; ++s) {
        if (s + 8 < S) {
            // gfx1250 global_prefetch_b8: stream randoms 8 spikes ahead
            __builtin_prefetch(rp + (s + 8) * HW, 0, 1);
        }
        const float rt = rp[s * HW] * total;   // cdf/total < r  <=>  cdf < r*total

        // Coarse level in registers: 7 VALU compares pick the 16-wide segment.
        int lo = 0;
#pragma unroll
        for (int j = 0; j < 7; ++j) lo += (coarse[j] < rt) ? 16 : 0;

        // 4-step binary search inside the segment (4 conflict-free ds reads).
#pragma unroll
        for (int st = 8; st >= 1; st >>= 1) {
            const float v = cdf[wbase + (lo + st - 1) * WPIX + lane];
            lo += (v < rt) ? st : 0;
        }
        // lo == #{c : cdf[c] < rt}; clamp like the reference's min(idx, C-1)
        const int idx = (lo > C - 1) ? (C - 1) : lo;
        op[s * HW] = (float)idx;
    }
}

} // namespace

extern "C" void kernel_launch(void* const* d_in, const int* in_sizes, int n_in,
                              void* d_out, int out_size, void* d_ws, size_t ws_size,
                              hipStream_t stream)
{
    const float* inp = (const float*)d_in[0];   // [B, C, H, W] f32
    const float* rnd = (const float*)d_in[1];   // [B, S, H, W] f32
    float*       out = (float*)d_out;           // [B, S, H, W] indices as f32

    const int npix   = in_sizes[0] / C;         // B*H*W = 262144
    const int blocks = npix / BLK;              // 2048

    spike_sample_kernel<<<blocks, BLK, 0, stream>>>(inp, rnd, out);

    (void)n_in; (void)out_size; (void)d_ws; (void)ws_size;
}